// GraphiT_Layer_82832739270777
// MI455X (gfx1250) — compile-verified
//
#include <hip/hip_runtime.h>
#include <hip/hip_bf16.h>

typedef __attribute__((ext_vector_type(2))) float v2f;
typedef __attribute__((ext_vector_type(8))) float v8f;

#define BATCH 2
#define NN    256
#define IND   128
#define NH    8
#define KD    64
#define HK    512   // NH*KD
#define BN    512   // BATCH*NN

// workspace layout (floats)
#define WS_Q   0
#define WS_K   (BN*HK)        // 262144
#define WS_V   (2*BN*HK)      // 524288
#define WS_QK  (3*BN*HK)      // 786432, size 2*8*256*256 = 1048576 floats

// ---------------------------------------------------------------------------
// Kernel A: Q/K/V projections via V_WMMA_F32_16X16X4_F32.
// One 16x16 output tile per wave; 3 GEMMs of [512x128]x[128x512].
// A layout (16x4 f32): lanes 0-15 hold rows M=0..15 with K={kk,kk+1},
//                      lanes 16-31 hold rows M=0..15 with K={kk+2,kk+3}.
// B layout (4x16 f32): VGPR0 = row K=kk (lanes0-15) / K=kk+2 (lanes16-31), etc.
// ---------------------------------------------------------------------------
__global__ void qkv_gemm_kernel(const float* __restrict__ hmat,
                                const float* __restrict__ Wq,
                                const float* __restrict__ Wk,
                                const float* __restrict__ Wv,
                                float* __restrict__ ws) {
    const int lane = threadIdx.x & 31;
    const int wid  = threadIdx.x >> 5;
    const int gw   = blockIdx.x * (blockDim.x >> 5) + wid;   // [0, 3072)
    const int mat  = gw >> 10;                               // 0=Q 1=K 2=V
    const int t    = gw & 1023;
    const int tm   = t >> 5, tn = t & 31;                    // 32x32 tiles of 16
    const float* W = (mat == 0) ? Wq : ((mat == 1) ? Wk : Wv);
    const float scale = (mat == 1) ? 0.125f : 1.0f;          // K^-0.5 on Km
    float* out = ws + (size_t)mat * (BN * HK);

    const int hi = lane >> 4;        // which K-pair
    const int lo = lane & 15;
    const float* arow = hmat + (size_t)(tm * 16 + lo) * IND + 2 * hi; // h row
    const float* brow = W    + (size_t)(tn * 16 + lo) * IND + 2 * hi; // W row (B = W^T)

    v8f c = {};
    for (int kk = 0; kk < IND; kk += 4) {
        v2f a = *(const v2f*)(arow + kk);
        v2f b = *(const v2f*)(brow + kk);
        c = __builtin_amdgcn_wmma_f32_16x16x4_f32(false, a, false, b,
                                                  (short)0, c, false, false);
    }
#pragma unroll
    for (int r = 0; r < 8; ++r) {
        const int m = tm * 16 + r + 8 * hi;   // D: VGPR r = rows r / r+8
        const int n = tn * 16 + lo;
        out[(size_t)m * HK + n] = c[r] * scale;
    }
}

// ---------------------------------------------------------------------------
// Kernel B: qk[b,h,i,j] = sum_k Q[b,i,h,k]*Km[b,j,h,k] via WMMA.
// Per (b,h): [256x64]x[64x256]; 16x16 tile per wave, 16 k-steps.
// ---------------------------------------------------------------------------
__global__ void qk_gemm_kernel(const float* __restrict__ ws, float* __restrict__ qk) {
    const float* Q  = ws + WS_Q;
    const float* Km = ws + WS_K;
    const int lane = threadIdx.x & 31;
    const int wid  = threadIdx.x >> 5;
    const int gw   = blockIdx.x * (blockDim.x >> 5) + wid;   // [0, 4096)
    const int bh   = gw >> 8;                                // b*8 + h, [0,16)
    const int t    = gw & 255;
    const int ti   = t >> 4, tj = t & 15;
    const int b    = bh >> 3, hh = bh & 7;

    const int hi = lane >> 4;
    const int lo = lane & 15;
    const float* qrow = Q  + (size_t)(b * NN + ti * 16 + lo) * HK + hh * KD + 2 * hi;
    const float* krow = Km + (size_t)(b * NN + tj * 16 + lo) * HK + hh * KD + 2 * hi;

    v8f c = {};
    for (int kk = 0; kk < KD; kk += 4) {
        v2f a = *(const v2f*)(qrow + kk);
        v2f bb = *(const v2f*)(krow + kk);
        c = __builtin_amdgcn_wmma_f32_16x16x4_f32(false, a, false, bb,
                                                  (short)0, c, false, false);
    }
    float* outb = qk + (size_t)bh * NN * NN;
#pragma unroll
    for (int r = 0; r < 8; ++r) {
        const int i = ti * 16 + r + 8 * hi;
        const int j = tj * 16 + lo;
        outb[(size_t)i * NN + j] = c[r];
    }
}

// ---------------------------------------------------------------------------
// Kernel C: streaming online softmax over j, fused V + e_value reduction.
// One wave per (b,i,h); each lane owns k = {2*lane, 2*lane+1}.
// e_att / e_value are 256 MB each, consumed exactly once -> non-temporal
// loads so they don't evict the hot set (V 1MB, qk 4MB) from the 192MB L2.
// out[b,i,h,k] = (sum_j p_j*(V[j,k]+e_value[j,k])) / (sum_j p_j)
// ---------------------------------------------------------------------------
__global__ void attn_stream_kernel(const float* __restrict__ e_att,
                                   const float* __restrict__ e_value,
                                   const unsigned char* __restrict__ mask,
                                   const float* __restrict__ ws,
                                   const float* __restrict__ qk,
                                   float* __restrict__ out) {
    const float* Vw = ws + WS_V;
    const int lane = threadIdx.x & 31;
    const int wid  = threadIdx.x >> 5;
    const int gw   = blockIdx.x * (blockDim.x >> 5) + wid;   // [0, 4096)
    const int hh   = gw & 7;
    const int bi   = gw >> 3;            // b*256 + i
    const int b    = bi >> 8;
    const int kk   = lane * 2;

    const float* ea = e_att   + (((size_t)bi * NN) * NH + hh) * KD + kk; // j stride 512
    const float* ev = e_value + (((size_t)bi * NN) * NH + hh) * KD + kk;
    const float* vp = Vw + (size_t)b * NN * HK + hh * KD + kk;           // j stride 512
    const float* qp = qk + ((size_t)(b * NH + hh) * NN + (bi & 255)) * NN;
    const unsigned char* mp = mask + (size_t)bi * NN;

    const float LOG2E = 1.44269504088896340736f;
    float m0 = -1e30f, m1 = -1e30f;
    float l0 = 0.0f, l1 = 0.0f;
    float a0 = 0.0f, a1 = 0.0f;

    for (int jc = 0; jc < NN; jc += 32) {
        const float qv = qp[jc + lane];
        const unsigned char mk = mp[jc + lane];
        const float sbase = (qv - (mk ? 0.0f : 1e24f)) * LOG2E;
        const float pmul  = mk ? 1.0f : 0.0f;
#pragma unroll 8
        for (int jj = 0; jj < 32; ++jj) {
            const float sb = __shfl(sbase, jj, 32);
            const float pm = __shfl(pmul, jj, 32);
            const size_t joff = (size_t)(jc + jj) * 512;
            const v2f eav = __builtin_nontemporal_load((const v2f*)(ea + joff));
            const v2f evv = __builtin_nontemporal_load((const v2f*)(ev + joff));
            const float2 vv = *(const float2*)(vp + joff);
            const float s0 = __builtin_fmaf(eav.x, LOG2E, sb);
            const float s1 = __builtin_fmaf(eav.y, LOG2E, sb);
            const float nm0 = fmaxf(m0, s0);
            const float nm1 = fmaxf(m1, s1);
            const float c0 = __builtin_amdgcn_exp2f(m0 - nm0);
            const float c1 = __builtin_amdgcn_exp2f(m1 - nm1);
            const float p0 = pm * __builtin_amdgcn_exp2f(s0 - nm0);
            const float p1 = pm * __builtin_amdgcn_exp2f(s1 - nm1);
            l0 = l0 * c0 + p0;
            l1 = l1 * c1 + p1;
            a0 = a0 * c0 + p0 * (vv.x + evv.x);
            a1 = a1 * c1 + p1 * (vv.y + evv.y);
            m0 = nm0;
            m1 = nm1;
        }
    }
    float* op = out + ((size_t)bi * NH + hh) * KD + kk;   // (b, n, H*K) layout
    v2f res;
    res.x = a0 / l0;
    res.y = a1 / l1;
    __builtin_nontemporal_store(res, (v2f*)op);
}

extern "C" void kernel_launch(void* const* d_in, const int* in_sizes, int n_in,
                              void* d_out, int out_size, void* d_ws, size_t ws_size,
                              hipStream_t stream) {
    const float* h      = (const float*)d_in[0];
    const float* e_att  = (const float*)d_in[1];
    const float* e_val  = (const float*)d_in[2];
    const unsigned char* mask = (const unsigned char*)d_in[3];
    const float* Wq     = (const float*)d_in[4];
    const float* Wk     = (const float*)d_in[5];
    const float* Wv     = (const float*)d_in[6];
    float* out = (float*)d_out;
    float* ws  = (float*)d_ws;
    float* qkp = ws + WS_QK;

    // A: 3 GEMMs * 1024 tiles = 3072 waves, 4 waves/block
    qkv_gemm_kernel<<<768, 128, 0, stream>>>(h, Wq, Wk, Wv, ws);
    // B: 16 (b,h) * 256 tiles = 4096 waves, 4 waves/block
    qk_gemm_kernel<<<1024, 128, 0, stream>>>(ws, qkp);
    // C: 4096 waves, 8 waves/block
    attn_stream_kernel<<<512, 256, 0, stream>>>(e_att, e_val, mask, ws, qkp, out);
}